// ConsistencyLoss_60954175865424
// MI455X (gfx1250) — compile-verified
//
#include <hip/hip_runtime.h>
#include <hip/hip_bf16.h>

#define NUM_MODES    6
#define FUTURE_STEPS 30
#define N_AGENTS_C   20000
#define NPERM        720
#define MAIN_BLOCKS  256
#define THREADS      256
#define WAVES_PER_BLOCK (THREADS / 32)

typedef float v2f __attribute__((ext_vector_type(2)));
typedef float v8f __attribute__((ext_vector_type(8)));

__device__ __forceinline__ float smooth_l1(float d) {
    float a = fabsf(d);
    return a < 1.0f ? 0.5f * d * d : a - 0.5f;
}

__global__ __launch_bounds__(THREADS)
void cl_main(const float* __restrict__ pred_past,
             const float* __restrict__ pred_now,
             const float* __restrict__ pad_loc,
             const unsigned char* __restrict__ mask,
             const float* __restrict__ target,
             float* __restrict__ partials) {
    __shared__ float ldsDist[WAVES_PER_BLOCK][40];  // 36 used per wave, padded
    __shared__ float ldsSum[WAVES_PER_BLOCK][3];

    const int lane = threadIdx.x & 31;
    const int w    = threadIdx.x >> 5;
    const int gw   = blockIdx.x * WAVES_PER_BLOCK + w;   // global wave id
    const int tw   = gridDim.x * WAVES_PER_BLOCK;        // total waves
    const int iters = (N_AGENTS_C + tw - 1) / tw;

    float accReg = 0.f, accCons = 0.f, accValid = 0.f;

    for (int it = 0; it < iters; ++it) {
        const int n = gw + it * tw;          // wave-uniform agent id
        if (n < N_AGENTS_C) {                // uniform branch: EXEC stays all-ones
            const float valid = mask[n] ? 0.f : 1.f;
            const float tx = target[2 * n], ty = target[2 * n + 1];

            // ---------- endpoint data -> dist^2 matrix via WMMA ----------
            // A (16x4 f32): row m = [px, py, |p|^2, 1]; lanes 0-15 hold K=0,1; 16-31 K=2,3
            // B (4x16 f32): col j = [-2qx, -2qy, 1, |q|^2]; same K halving across lanes
            // C[m][j] = |p|^2 + |q|^2 - 2 p.q = dist^2
            const int half = lane >> 4;
            const int m    = lane & 15;
            float ax = 0.f, ay = 0.f, bx = 0.f, by = 0.f;
            if (m < NUM_MODES) {
                const int baseEnd = ((m * N_AGENTS_C + n) * FUTURE_STEPS + (FUTURE_STEPS - 1)) * 4;
                const int basePad = (m * N_AGENTS_C + n) * 2;
                float px = pred_past[baseEnd]     + pad_loc[basePad];
                float py = pred_past[baseEnd + 1] + pad_loc[basePad + 1];
                float qx = pred_now[baseEnd]      + tx;
                float qy = pred_now[baseEnd + 1]  + ty;
                if (half == 0) { ax = px; ay = py; bx = -2.f * qx; by = -2.f * qy; }
                else           { ax = px * px + py * py; ay = 1.f; bx = 1.f; by = qx * qx + qy * qy; }
            }
#if __has_builtin(__builtin_amdgcn_wmma_f32_16x16x4_f32)
            v2f A; A.x = ax; A.y = ay;
            v2f B; B.x = bx; B.y = by;
            v8f C = {0.f, 0.f, 0.f, 0.f, 0.f, 0.f, 0.f, 0.f};
            C = __builtin_amdgcn_wmma_f32_16x16x4_f32(false, A, false, B, (short)0, C, false, false);
            if (lane < NUM_MODES) {
                #pragma unroll
                for (int r = 0; r < NUM_MODES; ++r)
                    ldsDist[w][r * NUM_MODES + lane] = sqrtf(fmaxf(C[r], 0.f));
            }
#else
            // VALU fallback: lane j<6 computes column j of the distance matrix
            if (lane < NUM_MODES) {
                const int j  = lane;
                const int bq = ((j * N_AGENTS_C + n) * FUTURE_STEPS + FUTURE_STEPS - 1) * 4;
                float qx = pred_now[bq] + tx, qy = pred_now[bq + 1] + ty;
                #pragma unroll
                for (int i = 0; i < NUM_MODES; ++i) {
                    const int bp = ((i * N_AGENTS_C + n) * FUTURE_STEPS + FUTURE_STEPS - 1) * 4;
                    const int pb = (i * N_AGENTS_C + n) * 2;
                    float px = pred_past[bp] + pad_loc[pb];
                    float py = pred_past[bp + 1] + pad_loc[pb + 1];
                    float dx = px - qx, dy = py - qy;
                    ldsDist[w][i * NUM_MODES + j] = sqrtf(dx * dx + dy * dy);
                }
            }
#endif
            __builtin_amdgcn_wave_barrier();      // LDS is in-order within a wave
            asm volatile("" ::: "memory");

            // ---------- brute-force 720-permutation assignment ----------
            // factoradic decode of perm index == itertools lexicographic order
            float best = 3.0e38f; int bestIdx = 0;
            for (int p = lane; p < NPERM; p += 32) {
                unsigned avail = 0x543210u;       // nibble list of remaining modes
                int k = p;
                float s = 0.f;
                #pragma unroll
                for (int i = 0; i < 6; ++i) {
                    const int fct[6] = {120, 24, 6, 2, 1, 1};
                    int d = k / fct[i]; k -= d * fct[i];
                    unsigned sh  = 4u * (unsigned)d;
                    unsigned sel = (avail >> sh) & 0xFu;
                    unsigned low = avail & ((1u << sh) - 1u);
                    avail = ((avail >> (sh + 4u)) << sh) | low;
                    s += ldsDist[w][i * NUM_MODES + (int)sel];
                }
                if (s < best) { best = s; bestIdx = p; }   // strict < keeps earliest
            }
            #pragma unroll
            for (int off = 16; off > 0; off >>= 1) {       // first-min tie-break
                float ov = __shfl_xor(best, off, 32);
                int   oi = __shfl_xor(bestIdx, off, 32);
                if (ov < best || (ov == best && oi < bestIdx)) { best = ov; bestIdx = oi; }
            }
            unsigned permPacked = 0;                        // decode winner -> nibbles
            {
                unsigned avail = 0x543210u;
                int k = bestIdx;
                #pragma unroll
                for (int i = 0; i < 6; ++i) {
                    const int fct[6] = {120, 24, 6, 2, 1, 1};
                    int d = k / fct[i]; k -= d * fct[i];
                    unsigned sh  = 4u * (unsigned)d;
                    unsigned sel = (avail >> sh) & 0xFu;
                    unsigned low = avail & ((1u << sh) - 1u);
                    avail = ((avail >> (sh + 4u)) << sh) | low;
                    permPacked |= sel << (4 * i);
                }
            }

            // ---------- consistency loss: stream 6x30x2 elements ----------
            float cons = 0.f;
            for (int idx = lane; idx < NUM_MODES * FUTURE_STEPS; idx += 32) {
                int i  = idx / FUTURE_STEPS;
                int t  = idx - i * FUTURE_STEPS;
                int pi = (int)((permPacked >> (4 * i)) & 0xFu);
                const float2 lp2 = *reinterpret_cast<const float2*>(
                    pred_past + ((i * N_AGENTS_C + n) * FUTURE_STEPS + t) * 4);
                const float2 pl2 = *reinterpret_cast<const float2*>(
                    pad_loc + (i * N_AGENTS_C + n) * 2);
                const float2 nw2 = *reinterpret_cast<const float2*>(
                    pred_now + ((pi * N_AGENTS_C + n) * FUTURE_STEPS + t) * 4);
                float lx = lp2.x + pl2.x, ly = lp2.y + pl2.y;
                float sx = nw2.x + tx,    sy = nw2.y + ty;
                cons += smooth_l1(lx - sx) + smooth_l1(ly - sy);
            }
            accCons += valid * cons;

            // ---------- pad_loc reg loss (12 terms) + valid count ----------
            if (lane < 2 * NUM_MODES) {
                int f = lane >> 1, c = lane & 1;
                float pv = pad_loc[(f * N_AGENTS_C + n) * 2 + c];
                float tv = c ? ty : tx;
                accReg += valid * smooth_l1(pv - tv);
            }
            if (lane == 0) accValid += valid;
        }
    }

    // wave reduction, then block reduction -> one partial per block
    #pragma unroll
    for (int off = 16; off > 0; off >>= 1) {
        accReg   += __shfl_xor(accReg,   off, 32);
        accCons  += __shfl_xor(accCons,  off, 32);
        accValid += __shfl_xor(accValid, off, 32);
    }
    if (lane == 0) { ldsSum[w][0] = accReg; ldsSum[w][1] = accCons; ldsSum[w][2] = accValid; }
    __syncthreads();
    if (threadIdx.x == 0) {
        float r = 0.f, c = 0.f, v = 0.f;
        for (int i = 0; i < WAVES_PER_BLOCK; ++i) { r += ldsSum[i][0]; c += ldsSum[i][1]; v += ldsSum[i][2]; }
        partials[blockIdx.x * 3 + 0] = r;
        partials[blockIdx.x * 3 + 1] = c;
        partials[blockIdx.x * 3 + 2] = v;
    }
}

__global__ __launch_bounds__(THREADS)
void cl_finalize(const float* __restrict__ partials, float* __restrict__ out, int nParts) {
    __shared__ float s0[THREADS], s1[THREADS], s2[THREADS];
    int t = threadIdx.x;
    float r = 0.f, c = 0.f, v = 0.f;
    if (t < nParts) { r = partials[t * 3]; c = partials[t * 3 + 1]; v = partials[t * 3 + 2]; }
    s0[t] = r; s1[t] = c; s2[t] = v;
    __syncthreads();
    for (int off = THREADS / 2; off > 0; off >>= 1) {
        if (t < off) { s0[t] += s0[t + off]; s1[t] += s1[t + off]; s2[t] += s2[t + off]; }
        __syncthreads();
    }
    if (t == 0) {
        float nv = fmaxf(s2[0], 1.0f);
        out[0] = s0[0] / (2.0f * (float)NUM_MODES * nv);
        out[1] = s1[0] / ((float)(NUM_MODES * FUTURE_STEPS * 2) * nv);
    }
}

extern "C" void kernel_launch(void* const* d_in, const int* in_sizes, int n_in,
                              void* d_out, int out_size, void* d_ws, size_t ws_size,
                              hipStream_t stream) {
    const float*         pred_past = (const float*)d_in[0];
    const float*         pred_now  = (const float*)d_in[1];
    const float*         pad_loc   = (const float*)d_in[2];
    const unsigned char* mask      = (const unsigned char*)d_in[3];
    const float*         target    = (const float*)d_in[4];
    float* partials = (float*)d_ws;   // MAIN_BLOCKS * 3 floats, fully written each call

    cl_main<<<MAIN_BLOCKS, THREADS, 0, stream>>>(pred_past, pred_now, pad_loc, mask, target, partials);
    cl_finalize<<<1, THREADS, 0, stream>>>(partials, (float*)d_out, MAIN_BLOCKS);
}